// MountainCar_7318624272632
// MI455X (gfx1250) — compile-verified
//
#include <hip/hip_runtime.h>

typedef float v2f __attribute__((ext_vector_type(2)));
typedef float v8f __attribute__((ext_vector_type(8)));

#define GOAL_P  0.5f
#define MIN_P   (-1.2f)
#define MIN_V   (-0.07f)
#define MAX_V   (0.07f)
#define MIN_ACC (-1.2f)
#define MAX_ACC (1.2f)
#define MAX_STEPS 500

// One wave32 simulates 16 envs. State (p,v,u,r) lives in lanes 0-15.
// Layer 1: h[16 envs x 64 hidden] via 4x V_WMMA_F32_16X16X4_F32 with bias fused:
//   A (16x4) rows = [p, v, 1, 0]; B_t (4x16) = [w1[0,tile]; w1[1,tile]; b1[tile]; 0]
// Layer 2: per-lane relu+fma vs w2 (compiler packs into v_pk_fma_f32), then a
//   reduce-scatter: 3 register-halving butterfly rounds (masks 1,2,4 splitting
//   the 8 row-registers on row bits 0..2 so row l&7 lands in lane l), one
//   same-row completion round (mask 8), one cross-half shuffle (mask 16).
__global__ __launch_bounds__(256)
void MountainCar_7318624272632_kernel(const float* __restrict__ x,
                                      const float* __restrict__ w1,
                                      const float* __restrict__ b1,
                                      const float* __restrict__ w2,
                                      const float* __restrict__ b2,
                                      float* __restrict__ out, int B)
{
    const int lane   = threadIdx.x & 31;
    const int waveId = (blockIdx.x * blockDim.x + threadIdx.x) >> 5;
    const int ln     = lane & 15;
    const bool hiHalf = (lane >= 16);

    if (waveId * 16 >= B) return;            // wave-uniform guard (EXEC stays full)

    const int  env      = waveId * 16 + ln;
    const bool envValid = (env < B);
    const int  envC     = envValid ? env : (B - 1);

    // ---- Load weights into VGPRs (once). B-matrix layout for 4x16 f32:
    // lanes 0-15 hold K=0 (VGPR0) / K=1 (VGPR1); lanes 16-31 hold K=2 / K=3.
    v2f  Bw[4];
    float w2l[4];
#pragma unroll
    for (int t = 0; t < 4; ++t) {
        const int col = t * 16 + ln;
        Bw[t][0] = hiHalf ? b1[col] : w1[col];        // K=0 row / K=2 (bias) row
        Bw[t][1] = hiHalf ? 0.0f    : w1[64 + col];   // K=1 row / K=3 (zero) row
        w2l[t]   = w2[col];
    }
    const float bias2 = b2[0];

    // ---- Load state (all lanes load a valid address; hi half duplicates lo half)
    const float4 st = reinterpret_cast<const float4*>(x)[envC];
    float p = st.x, v = st.y, u = st.z, r = st.w;

    const bool bit0 = (lane & 1) != 0;
    const bool bit1 = (lane & 2) != 0;
    const bool bit2 = (lane & 4) != 0;

    for (int step = 0; step < MAX_STEPS; ++step) {
        const bool active = (p <= GOAL_P);
        // Uniform early-out once every live env in this wave is frozen.
        if (__ballot(active && !hiHalf) == 0ull) break;

        const bool hit = (p <= MIN_P);
        const float p1 = hit ? MIN_P : p;
        const float v1 = hit ? 0.0f  : v;

        // A matrix: lanes 0-15 -> (p1, v1); lanes 16-31 -> (1, 0) for bias column
        v2f A;
        A[0] = hiHalf ? 1.0f : p1;
        A[1] = hiHalf ? 0.0f : v1;

        float acc[8];
#pragma unroll
        for (int i = 0; i < 8; ++i) acc[i] = 0.0f;

#pragma unroll
        for (int t = 0; t < 4; ++t) {
            v8f c = {};
            c = __builtin_amdgcn_wmma_f32_16x16x4_f32(
                    /*neg_a=*/false, A, /*neg_b=*/false, Bw[t],
                    /*c_mod=*/(short)0, c, /*reuse_a=*/false, /*reuse_b=*/false);
            // D layout: VGPR i, lane l -> h[row = i + 8*(l>=16)][col = 16*t + l%16]
#pragma unroll
            for (int i = 0; i < 8; ++i) {
                const float h = fmaxf(c[i], 0.0f);           // ReLU
                acc[i] = fmaf(h, w2l[t], acc[i]);            // * w2, sum over tiles
            }
        }

        // ---- Reduce-scatter: row (lane&7) of this half ends up at this lane.
        // Round 1 (mask 1): keep rows with bit0 == lane bit0.
        float r1a[4];
#pragma unroll
        for (int j = 0; j < 4; ++j) {
            const float keep = bit0 ? acc[2 * j + 1] : acc[2 * j];
            const float send = bit0 ? acc[2 * j]     : acc[2 * j + 1];
            r1a[j] = keep + __shfl_xor(send, 1, 32);
        }
        // Round 2 (mask 2): keep rows with bit1 == lane bit1.
        float r2a[2];
#pragma unroll
        for (int j = 0; j < 2; ++j) {
            const float keep = bit1 ? r1a[2 * j + 1] : r1a[2 * j];
            const float send = bit1 ? r1a[2 * j]     : r1a[2 * j + 1];
            r2a[j] = keep + __shfl_xor(send, 2, 32);
        }
        // Round 3 (mask 4): keep row with bit2 == lane bit2.
        {
            const float keep = bit2 ? r2a[1] : r2a[0];
            const float send = bit2 ? r2a[0] : r2a[1];
            r2a[0] = keep + __shfl_xor(send, 4, 32);
        }
        // Round 4 (mask 8): complete the 16-lane sum for this row.
        const float rowSum = r2a[0] + __shfl_xor(r2a[0], 8, 32);
        // Cross-half: lanes 0-15 fetch rows 8-15 (and vice versa).
        const float rowSumX = __shfl_xor(rowSum, 16, 32);

        const float upre = ((ln & 8) ? rowSumX : rowSum) + bias2;

        float u1 = tanhf(upre);
        u1 = fminf(fmaxf(u1, MIN_ACC), MAX_ACC);

        const float rr = fmaf(-0.1f * u1, u1, r);
        float v2 = v1 + 0.0015f * u1 - 0.0025f * __cosf(3.0f * p1);
        v2 = fminf(fmaxf(v2, MIN_V), MAX_V);
        const float p2 = p1 + v2;

        // state = where(active, new, state)  (branchless -> v_cndmask)
        p = active ? p2 : p;
        v = active ? v2 : v;
        u = active ? u1 : u;
        r = active ? rr : r;
    }

    const float rout = (p <= GOAL_P) ? r : (r + 100.0f);
    if (!hiHalf && envValid) {
        float4 o;
        o.x = p; o.y = v; o.z = u; o.w = rout;
        reinterpret_cast<float4*>(out)[env] = o;
    }
}

extern "C" void kernel_launch(void* const* d_in, const int* in_sizes, int n_in,
                              void* d_out, int out_size, void* d_ws, size_t ws_size,
                              hipStream_t stream) {
    const float* x  = (const float*)d_in[0];   // (B,4)
    const float* w1 = (const float*)d_in[1];   // (2,64)
    const float* b1 = (const float*)d_in[2];   // (64,)
    const float* w2 = (const float*)d_in[3];   // (64,1)
    const float* b2 = (const float*)d_in[4];   // (1,)
    float* out = (float*)d_out;                // (B,4)

    const int B = in_sizes[0] / 4;
    const int envsPerWave = 16;
    const int waves = (B + envsPerWave - 1) / envsPerWave;
    const int threads = waves * 32;
    const int block = 256;
    const int grid = (threads + block - 1) / block;

    MountainCar_7318624272632_kernel<<<grid, block, 0, stream>>>(x, w1, b1, w2, b2, out, B);
}